// Simple_edge_encoder_47278999994906
// MI455X (gfx1250) — compile-verified
//
#include <hip/hip_runtime.h>

// ---------------------------------------------------------------------------
// Simple_edge_encoder on MI455X (gfx1250, wave32, WMMA).
// Output = segment_sum(relu(BN(concat(h0[src], m) @ W0^T + b0)), dst)
// (2nd layer's GEMM/BN is dead code w.r.t. the returned value.)
// GEMM uses V_WMMA_F32_16X16X4_F32 (fp32-exact); the GEMM is recomputed in the
// BN-apply pass instead of spilling the 512MB x0 intermediate to HBM
// (trades ~26us of WMMA flops for ~45us of HBM traffic at 23.3 TB/s).
// All global f32 accumulations use hardware global_atomic_add_f32 via inline
// asm (scope:SCOPE_DEV) so the backend cannot fall back to a CAS loop; both
// accumulation targets (25.6MB h0, 51.2MB h1) are resident in the 192MB L2.
// ---------------------------------------------------------------------------

typedef __attribute__((ext_vector_type(2))) float v2f;
typedef __attribute__((ext_vector_type(8))) float v8f;

#define D0 64            // input feature dim
#define D1 128           // layer-0 output dim (= 2*D0 concat input dim)
#define TPB 4            // M-tiles (waves) per block
#define BLOCK_THREADS 128
#define A_STRIDE 132     // 128 + 4 pad: conflict-free LDS, keeps 16B alignment

// Hardware no-return f32 atomic add at device scope. No-return form is
// STOREcnt-tracked; drained by the implicit wait-idle at S_ENDPGM.
__device__ __forceinline__ void atomic_add_dev(float* p, float v) {
    asm volatile("global_atomic_add_f32 %0, %1, off scope:SCOPE_DEV"
                 :: "v"(p), "v"(v) : "memory");
}

// -------------------- zero workspace + output ------------------------------
__global__ void k_zero(float4* a, long na4, float4* b, long nb4) {
    long i = (long)blockIdx.x * blockDim.x + threadIdx.x;
    long stride = (long)gridDim.x * blockDim.x;
    const float4 z = {0.0f, 0.0f, 0.0f, 0.0f};
    for (long j = i; j < na4; j += stride) a[j] = z;
    for (long j = i; j < nb4; j += stride) b[j] = z;
}

// -------------------- h0 = segment_sum(m, dst) -----------------------------
// one thread per 4 channels of one edge; coalesced float4 loads, f32 atomics
// land in L2 (h0 is 25.6MB, resident in 192MB L2).
__global__ void k_scatter0(const float* __restrict__ m, const int* __restrict__ dst,
                           float* __restrict__ h0, int E) {
    int t = blockIdx.x * blockDim.x + threadIdx.x;
    if (t >= E * 16) return;
    int e = t >> 4;
    int c = (t & 15) << 2;
    const float4 v = *(const float4*)(m + (long)e * D0 + c);
    float* p = h0 + (long)dst[e] * D0 + c;
    atomic_add_dev(p + 0, v.x);
    atomic_add_dev(p + 1, v.y);
    atomic_add_dev(p + 2, v.z);
    atomic_add_dev(p + 3, v.w);
}

// -------------------- shared GEMM pieces -----------------------------------
// Stage one 16-edge x 128-feature A tile: row e = [ h0[src[e]] (64) | m[e] (64) ]
__device__ __forceinline__ void stage_tile(float* At, const float* __restrict__ h0,
                                           const float* __restrict__ m,
                                           const int* __restrict__ src,
                                           long tb, int E, int lane) {
    int row  = lane >> 1;                 // edge within tile (0..15), 2 lanes/edge
    int half = lane & 1;                  // 0: gathered h0[src], 1: m
    long e = tb + row; if (e >= E) e = E - 1;
    const float* sp = half ? (m + (long)e * D0) : (h0 + (long)src[e] * D0);
    float* dp = At + row * A_STRIDE + half * 64;
#pragma unroll
    for (int i = 0; i < 16; ++i)
        *(float4*)(dp + i * 4) = *(const float4*)(sp + i * 4);
}

// One 16x16 output tile (channels nt*16..+15) for this wave's 16 edges.
// Per-lane WMMA operands (wave32, V_WMMA_F32_16X16X4_F32):
//   A vgpr j, lane l -> A[l&15, 4k + 2*(l>>4) + j]
//   B vgpr j, lane l -> B[4k + 2*(l>>4) + j, l&15] = W0[nt*16 + (l&15), ...]
//   D vgpr r, lane l -> x0[r + 8*(l>>4), nt*16 + (l&15)]
__device__ __forceinline__ v8f gemm_ntile(const float* At, const float* __restrict__ W,
                                          const float* __restrict__ b, int nt, int lane) {
    int nlane = lane & 15;
    int kb = (lane >> 4) << 1;            // 0 or 2
    float bb = b[nt * 16 + nlane];
    v8f c = {bb, bb, bb, bb, bb, bb, bb, bb};
    const float* wrow = W + (long)(nt * 16 + nlane) * (2 * D0) + kb;  // W0 row-major [128][128]
    const float* arow = At + nlane * A_STRIDE + kb;
#pragma unroll
    for (int k = 0; k < 32; ++k) {        // K = 128 in steps of 4
        v2f a = *(const v2f*)(arow + 4 * k);
        v2f w = *(const v2f*)(wrow + 4 * k);
        c = __builtin_amdgcn_wmma_f32_16x16x4_f32(false, a, false, w, (short)0, c,
                                                  false, false);
    }
    return c;
}

// -------------------- pass 1: GEMM + BN statistics -------------------------
__global__ void __launch_bounds__(BLOCK_THREADS)
k_gemm_stats(const float* __restrict__ h0, const float* __restrict__ m,
             const int* __restrict__ src, const float* __restrict__ W0,
             const float* __restrict__ b0, float* __restrict__ gsum,
             float* __restrict__ gsq, int E) {
    __shared__ float At[TPB][16 * A_STRIDE];
    __shared__ float bsum[D1], bsq[D1];
    int tid = threadIdx.x, lane = tid & 31, wid = tid >> 5;
    bsum[tid] = 0.0f; bsq[tid] = 0.0f;    // BLOCK_THREADS == D1
    long tb = ((long)blockIdx.x * TPB + wid) * 16;
    stage_tile(At[wid], h0, m, src, tb, E, lane);
    __syncthreads();
#pragma unroll
    for (int nt = 0; nt < 8; ++nt) {
        v8f c = gemm_ntile(At[wid], W0, b0, nt, lane);
        float s = 0.0f, q = 0.0f;
#pragma unroll
        for (int r = 0; r < 8; ++r) { float x = c[r]; s += x; q += x * x; }
        int ch = nt * 16 + (lane & 15);   // lanes l and l+16 share a channel
        (void)__hip_atomic_fetch_add(&bsum[ch], s, __ATOMIC_RELAXED, __HIP_MEMORY_SCOPE_WORKGROUP);
        (void)__hip_atomic_fetch_add(&bsq[ch],  q, __ATOMIC_RELAXED, __HIP_MEMORY_SCOPE_WORKGROUP);
    }
    __syncthreads();
    atomic_add_dev(&gsum[tid], bsum[tid]);
    atomic_add_dev(&gsq[tid],  bsq[tid]);
}

// -------------------- fold BN into per-channel scale/shift -----------------
__global__ void k_finalize(const float* gsum, const float* gsq, const float* g,
                           const float* beta, float* scale, float* shift, float invE) {
    int t = threadIdx.x;                  // 128 threads
    float mu  = gsum[t] * invE;
    float var = gsq[t] * invE - mu * mu;  // biased var, as in reference
    float inv = rsqrtf(var + 1e-5f);
    float sc  = g[t] * inv;
    scale[t] = sc;
    shift[t] = beta[t] - mu * sc;
}

// -------------------- pass 2: GEMM + BN/relu + scatter to output -----------
__global__ void __launch_bounds__(BLOCK_THREADS)
k_gemm_scatter(const float* __restrict__ h0, const float* __restrict__ m,
               const int* __restrict__ src, const int* __restrict__ dst,
               const float* __restrict__ W0, const float* __restrict__ b0,
               const float* __restrict__ scale, const float* __restrict__ shift,
               float* __restrict__ out, int E) {
    __shared__ float At[TPB][16 * A_STRIDE];
    int tid = threadIdx.x, lane = tid & 31, wid = tid >> 5;
    long tb = ((long)blockIdx.x * TPB + wid) * 16;
    stage_tile(At[wid], h0, m, src, tb, E, lane);
    int drow[8]; bool valid[8];
#pragma unroll
    for (int r = 0; r < 8; ++r) {         // edge for D-vgpr r on this lane
        long e = tb + r + ((lane >> 4) << 3);
        valid[r] = (e < E);
        drow[r] = dst[valid[r] ? e : (E - 1)];
    }
    __syncthreads();
#pragma unroll
    for (int nt = 0; nt < 8; ++nt) {
        v8f c = gemm_ntile(At[wid], W0, b0, nt, lane);
        int ch = nt * 16 + (lane & 15);
        float sc = scale[ch], sh = shift[ch];
#pragma unroll
        for (int r = 0; r < 8; ++r) {
            float y = fmaxf(c[r] * sc + sh, 0.0f);
            if (valid[r])                 // lanes 0-15 hit consecutive addresses
                atomic_add_dev(&out[(long)drow[r] * D1 + ch], y);
        }
    }
}

// ---------------------------------------------------------------------------
extern "C" void kernel_launch(void* const* d_in, const int* in_sizes, int n_in,
                              void* d_out, int out_size, void* d_ws, size_t ws_size,
                              hipStream_t stream) {
    const int E = in_sizes[0] / D0;       // m is [E, 64]
    const int N = out_size / D1;          // output h1 is [N, 128]

    const float* m     = (const float*)d_in[0];
    const float* W0    = (const float*)d_in[1];
    const float* b0    = (const float*)d_in[2];
    const float* g0    = (const float*)d_in[3];
    const float* beta0 = (const float*)d_in[4];
    // d_in[5..8] (W1,b1,g1,beta1) are dead code for the returned value.
    const int* src = (const int*)d_in[9];
    const int* dst = (const int*)d_in[10];
    float* out = (float*)d_out;

    // workspace: h0 [N*64] | gsum[128] | gsq[128] | scale[128] | shift[128]
    float* h0    = (float*)d_ws;
    float* gsum  = h0 + (long)N * D0;
    float* gsq   = gsum + D1;
    float* scale = gsq + D1;
    float* shift = scale + D1;

    const long nz_ws4  = ((long)N * D0 + 2 * D1) / 4;  // all multiples of 4
    const long nz_out4 = ((long)N * D1) / 4;
    k_zero<<<2048, 256, 0, stream>>>((float4*)h0, nz_ws4, (float4*)out, nz_out4);

    const long t0 = (long)E * 16;
    k_scatter0<<<(int)((t0 + 255) / 256), 256, 0, stream>>>(m, dst, h0, E);

    const int ntiles = (E + 15) / 16;
    const int nblk = (ntiles + TPB - 1) / TPB;
    k_gemm_stats<<<nblk, BLOCK_THREADS, 0, stream>>>(h0, m, src, W0, b0, gsum, gsq, E);
    k_finalize<<<1, D1, 0, stream>>>(gsum, gsq, g0, beta0, scale, shift, 1.0f / (float)E);
    k_gemm_scatter<<<nblk, BLOCK_THREADS, 0, stream>>>(h0, m, src, dst, W0, b0,
                                                       scale, shift, out, E);
}